// RecommendationModel_75368086110268
// MI455X (gfx1250) — compile-verified
//
#include <hip/hip_runtime.h>
#include <hip/hip_bf16.h>

// ---------------------------------------------------------------------------
// RecommendationModel on MI455X (gfx1250, wave32).
// Dense projections use the native fp32 WMMA (V_WMMA_F32_16X16X4_F32): the
// workload is HBM/atomic bound (~17 GFLOP vs ~0.7 GB traffic @ 23.3 TB/s),
// so full-precision f32 matrix ops are the right ceiling-free choice.
// ---------------------------------------------------------------------------

#define H 128
#define N_USERS   100000
#define N_RECIPES 100000
#define N_ING_TAB 20000
#define N_ING     200000
#define E_ING     500000
#define E_UR      600000
#define F_REC     20

typedef float v2f __attribute__((ext_vector_type(2)));
typedef float v8f __attribute__((ext_vector_type(8)));

// ---------------- helpers ----------------

__device__ __forceinline__ void atomicMaxF(float* addr, float val) {
    // ordered-float atomic max via int/uint monotonic mapping
    if (val >= 0.0f) atomicMax((int*)addr, __float_as_int(val));
    else             atomicMin((unsigned int*)addr, __float_as_uint(val));
}

__device__ __forceinline__ float waveSum(float s) {
    #pragma unroll
    for (int off = 16; off; off >>= 1) s += __shfl_xor(s, off, 32);
    return s;
}

// ---------------- small utility kernels ----------------

__global__ void fill_f32(float* __restrict__ p, float v, int n) {
    int i = blockIdx.x * blockDim.x + threadIdx.x;
    if (i < n) p[i] = v;
}

// out[row*H + c] = bias[c]  (seed accumulator with bias)
__global__ void broadcast_bias(float* __restrict__ out, const float* __restrict__ b, int nrows) {
    int i = blockIdx.x * blockDim.x + threadIdx.x;
    if (i < nrows * H) out[i] = b[i & (H - 1)];
}

// r[row*H + c] += b[c]
__global__ void add_bias_inplace(float* __restrict__ r, const float* __restrict__ b, int nrows) {
    int i = blockIdx.x * blockDim.x + threadIdx.x;
    if (i < nrows * H) r[i] += b[i & (H - 1)];
}

// ---------------- gathers ----------------

// u[row] = user_table[id[row]] * min(1, 1/max(||.||,1e-7)); one wave per row.
__global__ void user_gather_renorm(const float* __restrict__ table, const int* __restrict__ ids,
                                   float* __restrict__ out, int n) {
    int row  = (blockIdx.x * blockDim.x + threadIdx.x) >> 5;
    int lane = threadIdx.x & 31;
    if (row >= n) return;
    const float4* src = (const float4*)(table + (size_t)ids[row] * H);
    float4 x = src[lane];
    float ss = waveSum(x.x * x.x + x.y * x.y + x.z * x.z + x.w * x.w);
    float nrm = sqrtf(ss);
    float sc  = fminf(1.0f, 1.0f / fmaxf(nrm, 1e-7f));
    float4 y = make_float4(x.x * sc, x.y * sc, x.z * sc, x.w * sc);
    ((float4*)(out + (size_t)row * H))[lane] = y;
}

// ing_x[row] = ing_table[id[row]]; one wave per row (float4 per lane).
__global__ void ing_gather(const float* __restrict__ table, const int* __restrict__ ids,
                           float* __restrict__ out, int n) {
    int row  = (blockIdx.x * blockDim.x + threadIdx.x) >> 5;
    int lane = threadIdx.x & 31;
    if (row >= n) return;
    const float4* src = (const float4*)(table + (size_t)ids[row] * H);
    ((float4*)(out + (size_t)row * H))[lane] = src[lane];
}

// ---------------- fp32 WMMA GEMM: C[N x 128] = A[N x K] @ W[K x 128] (+bias) --
// One wave owns a 16x16 output tile; 8 waves per block cover 16 rows x 128 cols.
// Fragment layouts per CDNA5 ISA 7.12.2 (32-bit A 16x4 / B 4x16 / C 16x16).

__launch_bounds__(256)
__global__ void gemm_wmma_f32(const float* __restrict__ A, const float* __restrict__ W,
                              const float* __restrict__ bias, float* __restrict__ C,
                              int N, int K) {
    const int wave = threadIdx.x >> 5;          // 0..7 -> column tile
    const int lane = threadIdx.x & 31;
    const int row0 = blockIdx.x * 16;
    const int col0 = wave * 16;
    if (row0 >= N) return;

    const int m    = lane & 15;                  // row (A) / col (B,C) within tile
    const int kb   = (lane >> 4) << 1;           // 0 for lanes 0-15, 2 for 16-31
    const int half = lane >> 4;

    v8f acc = {};
    for (int k = 0; k < K; k += 4) {
        v2f a, b;
        const float* ap = A + (size_t)(row0 + m) * K + (k + kb);
        a.x = ap[0];                             // A[m][k+kb]
        a.y = ap[1];                             // A[m][k+kb+1]
        const float* bp = W + (size_t)(k + kb) * H + col0 + m;
        b.x = bp[0];                             // W[k+kb  ][col]
        b.y = bp[H];                             // W[k+kb+1][col]
        acc = __builtin_amdgcn_wmma_f32_16x16x4_f32(
            /*neg_a=*/false, a, /*neg_b=*/false, b,
            /*c_mod=*/(short)0, acc, /*reuse_a=*/false, /*reuse_b=*/false);
    }

    float bv = bias ? bias[col0 + m] : 0.0f;
    #pragma unroll
    for (int v = 0; v < 8; ++v) {
        int row = row0 + half * 8 + v;           // lanes 0-15: M=v, 16-31: M=v+8
        C[(size_t)row * H + col0 + m] = acc[v] + bv;
    }
}

// ---------------- BatchNorm (training batch stats) ----------------

__global__ void bn_stats(const float* __restrict__ r, float* __restrict__ sum,
                         float* __restrict__ sumsq, int nrows) {
    int c = threadIdx.x;                         // 128 threads = one column each
    float s = 0.0f, s2 = 0.0f;
    for (int row = blockIdx.x; row < nrows; row += gridDim.x) {
        float v = r[(size_t)row * H + c];
        s += v; s2 += v * v;
    }
    atomicAdd(&sum[c], s);
    atomicAdd(&sumsq[c], s2);
}

__global__ void bn_finalize(const float* __restrict__ sum, const float* __restrict__ sumsq,
                            const float* __restrict__ gamma, const float* __restrict__ beta,
                            float* __restrict__ scale, float* __restrict__ shift, int nrows) {
    int c = threadIdx.x;
    if (c >= H) return;
    float inv_n = 1.0f / (float)nrows;
    float mean  = sum[c] * inv_n;
    float var   = sumsq[c] * inv_n - mean * mean;
    float sc    = gamma[c] * rsqrtf(var + 1e-5f);
    scale[c] = sc;
    shift[c] = beta[c] - mean * sc;
}

__global__ void bn_apply(float* __restrict__ r, const float* __restrict__ scale,
                         const float* __restrict__ shift, int nrows) {
    int i = blockIdx.x * blockDim.x + threadIdx.x;
    if (i < nrows * H) {
        int c = i & (H - 1);
        r[i] = r[i] * scale[c] + shift[c];
    }
}

// ---------------- GAT edge passes ----------------

// es[row] = dot(X[row], a); one wave per row
__global__ void rowdot(const float* __restrict__ X, const float* __restrict__ a,
                       float* __restrict__ out, int n) {
    int row  = (blockIdx.x * blockDim.x + threadIdx.x) >> 5;
    int lane = threadIdx.x & 31;
    if (row >= n) return;
    float4 x = ((const float4*)(X + (size_t)row * H))[lane];
    float4 av = ((const float4*)a)[lane];
    float s = waveSum(x.x * av.x + x.y * av.y + x.z * av.z + x.w * av.w);
    if (lane == 0) out[row] = s;
}

__global__ void edge_score_max(const float* __restrict__ es, const float* __restrict__ ed,
                               const int* __restrict__ src, const int* __restrict__ dst,
                               float* __restrict__ ebuf, float* __restrict__ segmax, int E) {
    int i = blockIdx.x * blockDim.x + threadIdx.x;
    if (i >= E) return;
    float e = es[src[i]] + ed[dst[i]];
    e = (e > 0.0f) ? e : 0.2f * e;               // leaky_relu(0.2)
    ebuf[i] = e;
    atomicMaxF(&segmax[dst[i]], e);
}

__global__ void edge_exp_den(const int* __restrict__ dst, const float* __restrict__ segmax,
                             float* __restrict__ ebuf, float* __restrict__ den, int E) {
    int i = blockIdx.x * blockDim.x + threadIdx.x;
    if (i >= E) return;
    int d = dst[i];
    float m = segmax[d];
    if (!(m > -3.3e38f)) m = 0.0f;               // empty-segment -inf -> 0
    float ex = expf(ebuf[i] - m);
    ebuf[i] = ex;
    atomicAdd(&den[d], ex);
}

// one wave per edge: out[dst] += (ex/den) * hs[src]
__global__ void edge_scatter(const float* __restrict__ ebuf, const float* __restrict__ den,
                             const int* __restrict__ src, const int* __restrict__ dst,
                             const float* __restrict__ hs, float* __restrict__ out, int E) {
    int i    = (blockIdx.x * blockDim.x + threadIdx.x) >> 5;
    int lane = threadIdx.x & 31;
    if (i >= E) return;
    int s = src[i], d = dst[i];
    float w = ebuf[i] / fmaxf(den[d], 1e-16f);
    float4 h = ((const float4*)(hs + (size_t)s * H))[lane];
    float* op = out + (size_t)d * H + lane * 4;
    atomicAdd(op + 0, w * h.x);
    atomicAdd(op + 1, w * h.y);
    atomicAdd(op + 2, w * h.z);
    atomicAdd(op + 3, w * h.w);
}

// ---------------- launch ----------------

extern "C" void kernel_launch(void* const* d_in, const int* in_sizes, int n_in,
                              void* d_out, int out_size, void* d_ws, size_t ws_size,
                              hipStream_t stream) {
    // setup_inputs() dict order
    const int*   user_id   = (const int*)d_in[0];
    const int*   ing_id    = (const int*)d_in[1];
    const float* rec_feat  = (const float*)d_in[2];
    /* recipe_id d_in[3] feeds an unused attribute */
    const int*   ing_src   = (const int*)d_in[4];
    const int*   ing_dst   = (const int*)d_in[5];
    const int*   ur_src    = (const int*)d_in[6];
    const int*   ur_dst    = (const int*)d_in[7];
    const float* user_tab  = (const float*)d_in[8];
    const float* ing_tab   = (const float*)d_in[9];
    /* img_table d_in[10] unused by output */
    const float* W_rec     = (const float*)d_in[11];
    const float* b_rec     = (const float*)d_in[12];
    const float* bn_gamma  = (const float*)d_in[13];
    const float* bn_beta   = (const float*)d_in[14];
    const float* g1_Ws     = (const float*)d_in[15];
    const float* g1_Wd     = (const float*)d_in[16];
    const float* g1_as     = (const float*)d_in[17];
    const float* g1_ad     = (const float*)d_in[18];
    const float* g1_b      = (const float*)d_in[19];
    const float* g2_Ws     = (const float*)d_in[20];
    const float* g2_Wd     = (const float*)d_in[21];
    const float* g2_as     = (const float*)d_in[22];
    const float* g2_ad     = (const float*)d_in[23];
    const float* g2_b      = (const float*)d_in[24];

    float* ws = (float*)d_ws;
    float* out = (float*)d_out;

    // workspace layout (floats); total ~90.7M floats (~363 MB)
    float* u      = ws;                             // 100000*128
    float* ing_x  = ws + 12800000;                  // 200000*128
    float* r      = ws + 38400000;                  // 100000*128
    float* hs     = ws + 51200000;                  // up to 200000*128
    float* hd     = ws + 76800000;                  // 100000*128
    float* es     = ws + 89600000;                  // up to 200000
    float* ed     = ws + 89800000;                  // 100000
    float* ebuf   = ws + 89900000;                  // up to 600000
    float* segmax = ws + 90500000;                  // 100000
    float* den    = ws + 90600000;                  // 100000
    float* bn_sum = ws + 90700000;                  // 128
    float* bn_sq  = bn_sum + 128;
    float* bn_sc  = bn_sum + 256;
    float* bn_sh  = bn_sum + 384;

    const int B = 256;
    const float NEG_INF = -__builtin_inff();

    // 1) user embedding with max_norm renorm; ingredient gather
    user_gather_renorm<<<(N_USERS * 32 + B - 1) / B, B, 0, stream>>>(user_tab, user_id, u, N_USERS);
    ing_gather<<<(N_ING * 32 + B - 1) / B, B, 0, stream>>>(ing_tab, ing_id, ing_x, N_ING);

    // 2) recipe linear (K=20, fp32 WMMA) + BatchNorm(batch stats)
    gemm_wmma_f32<<<N_RECIPES / 16, B, 0, stream>>>(rec_feat, W_rec, b_rec, r, N_RECIPES, F_REC);
    fill_f32<<<1, B, 0, stream>>>(bn_sum, 0.0f, 256);           // sum + sumsq
    bn_stats<<<512, H, 0, stream>>>(r, bn_sum, bn_sq, N_RECIPES);
    bn_finalize<<<1, H, 0, stream>>>(bn_sum, bn_sq, bn_gamma, bn_beta, bn_sc, bn_sh, N_RECIPES);
    bn_apply<<<(N_RECIPES * H + B - 1) / B, B, 0, stream>>>(r, bn_sc, bn_sh, N_RECIPES);

    // 3) GAT1: ingredient -> recipe, residual
    gemm_wmma_f32<<<N_ING / 16, B, 0, stream>>>(ing_x, g1_Ws, nullptr, hs, N_ING, H);
    gemm_wmma_f32<<<N_RECIPES / 16, B, 0, stream>>>(r, g1_Wd, nullptr, hd, N_RECIPES, H);
    rowdot<<<(N_ING * 32 + B - 1) / B, B, 0, stream>>>(hs, g1_as, es, N_ING);
    rowdot<<<(N_RECIPES * 32 + B - 1) / B, B, 0, stream>>>(hd, g1_ad, ed, N_RECIPES);
    fill_f32<<<(N_RECIPES + B - 1) / B, B, 0, stream>>>(segmax, NEG_INF, N_RECIPES);
    fill_f32<<<(N_RECIPES + B - 1) / B, B, 0, stream>>>(den, 0.0f, N_RECIPES);
    edge_score_max<<<(E_ING + B - 1) / B, B, 0, stream>>>(es, ed, ing_src, ing_dst, ebuf, segmax, E_ING);
    edge_exp_den<<<(E_ING + B - 1) / B, B, 0, stream>>>(ing_dst, segmax, ebuf, den, E_ING);
    // residual base: r <- r + g1_b, then scatter-add attention messages into r
    add_bias_inplace<<<(N_RECIPES * H + B - 1) / B, B, 0, stream>>>(r, g1_b, N_RECIPES);
    edge_scatter<<<(E_ING * 32 + B - 1) / B, B, 0, stream>>>(ebuf, den, ing_src, ing_dst, hs, r, E_ING);

    // 4) GAT2: user -> recipe (final output)
    gemm_wmma_f32<<<N_USERS / 16, B, 0, stream>>>(u, g2_Ws, nullptr, hs, N_USERS, H);
    gemm_wmma_f32<<<N_RECIPES / 16, B, 0, stream>>>(r, g2_Wd, nullptr, hd, N_RECIPES, H);
    rowdot<<<(N_USERS * 32 + B - 1) / B, B, 0, stream>>>(hs, g2_as, es, N_USERS);
    rowdot<<<(N_RECIPES * 32 + B - 1) / B, B, 0, stream>>>(hd, g2_ad, ed, N_RECIPES);
    fill_f32<<<(N_RECIPES + B - 1) / B, B, 0, stream>>>(segmax, NEG_INF, N_RECIPES);
    fill_f32<<<(N_RECIPES + B - 1) / B, B, 0, stream>>>(den, 0.0f, N_RECIPES);
    edge_score_max<<<(E_UR + B - 1) / B, B, 0, stream>>>(es, ed, ur_src, ur_dst, ebuf, segmax, E_UR);
    edge_exp_den<<<(E_UR + B - 1) / B, B, 0, stream>>>(ur_dst, segmax, ebuf, den, E_UR);
    broadcast_bias<<<(N_RECIPES * H + B - 1) / B, B, 0, stream>>>(out, g2_b, N_RECIPES);
    edge_scatter<<<(E_UR * 32 + B - 1) / B, B, 0, stream>>>(ebuf, den, ur_src, ur_dst, hs, out, E_UR);
}